// FourierLayer_7438883356766
// MI455X (gfx1250) — compile-verified
//
#include <hip/hip_runtime.h>
#include <math.h>

// Problem constants from the reference setup.
#define B_   32
#define T_   512
#define D_   256
#define P_   96
#define TP_  608          // T + pred_len
#define NF_  256          // freq rows 0..255 (row 0 masked; Nyquist never computed)
#define K_   16

typedef float v2f __attribute__((ext_vector_type(2)));
typedef float v8f __attribute__((ext_vector_type(8)));

// ---------------------------------------------------------------------------
// Packed trig tables (WMMA-ready: each entry holds the {K, K+1} pair a lane
// feeds into the 16x16x4 A-fragment, so one b64 load per operand).
// Analysis:  Wc/Ws [t2][f]   (v2f = {cos(2pi f(2t2)/512), cos(2pi f(2t2+1)/512)})
// Synthesis: Wc/Ws [f2][tau] (v2f = {cos(2pi(2f2)tau/512), cos(2pi(2f2+1)tau/512)})
// Exact integer phase reduction: angle = pi * ((f*t) & 511) / 256.
// ---------------------------------------------------------------------------
__global__ void build_dft_tables(v2f* __restrict__ Wc, v2f* __restrict__ Ws) {
    int idx = blockIdx.x * blockDim.x + threadIdx.x;      // (T/2)*NF = 65536
    if (idx >= (T_ / 2) * NF_) return;
    int t2 = idx >> 8;
    int f  = idx & 255;
    float s0, c0, s1, c1;
    sincospif((float)((f * (2 * t2    )) & (T_ - 1)) * (1.0f / 256.0f), &s0, &c0);
    sincospif((float)((f * (2 * t2 + 1)) & (T_ - 1)) * (1.0f / 256.0f), &s1, &c1);
    Wc[idx] = (v2f){c0, c1};
    Ws[idx] = (v2f){s0, s1};
}

__global__ void build_syn_tables(v2f* __restrict__ Wc, v2f* __restrict__ Ws) {
    int idx = blockIdx.x * blockDim.x + threadIdx.x;      // (NF/2)*TP = 77824
    if (idx >= (NF_ / 2) * TP_) return;
    int f2  = idx / TP_;
    int tau = idx - f2 * TP_;
    float s0, c0, s1, c1;
    sincospif((float)(((2 * f2    ) * tau) & (T_ - 1)) * (1.0f / 256.0f), &s0, &c0);
    sincospif((float)(((2 * f2 + 1) * tau) & (T_ - 1)) * (1.0f / 256.0f), &s1, &c1);
    Wc[idx] = (v2f){c0, c1};
    Ws[idx] = (v2f){s0, s1};
}

// Pack x (b,t,d) -> xp[b][t2][d] = {x[b][2t2][d], x[b][2t2+1][d]} (one pass, cheap).
__global__ void pack_x(const float* __restrict__ x, v2f* __restrict__ xp) {
    int idx = blockIdx.x * blockDim.x + threadIdx.x;      // B*(T/2)*D = 2097152
    if (idx >= B_ * (T_ / 2) * D_) return;
    int b  = idx >> 16;
    int t2 = (idx >> 8) & 255;
    int d  = idx & 255;
    const float* xr = x + ((size_t)b * T_ + 2 * t2) * D_ + d;
    xp[idx] = (v2f){xr[0], xr[D_]};
}

// ---------------------------------------------------------------------------
// Stage 1: DFT as f32 WMMA GEMM. One wave owns a 32(f) x 64(d) macro-tile
// (2x4 sub-tiles, Re+Im accumulators). Per K=4 step: 4 A b64 + 4 B b64 loads
// feed 16 v_wmma_f32_16x16x4_f32.
// Grid: 32 b * 8 fgroups * 4 dgroups = 1024 waves -> 256 blocks of 4 waves.
// ---------------------------------------------------------------------------
__global__ __launch_bounds__(128) void dft_wmma(
    const v2f* __restrict__ xp,     // [b][t2][d]
    const v2f* __restrict__ Wc,     // [t2][f]
    const v2f* __restrict__ Ws,
    float* __restrict__ Re, float* __restrict__ Im) {

    const int wave = (blockIdx.x << 2) + (threadIdx.x >> 5);   // 0..1023
    const int lane = threadIdx.x & 31;
    const int dg = wave & 3;
    const int fg = (wave >> 2) & 7;
    const int b  = wave >> 5;

    const int l15 = lane & 15;
    const int hk  = lane >> 4;          // half-wave K-pair select (K0,1 / K2,3)

    int fA[2], dB[4];
    #pragma unroll
    for (int i = 0; i < 2; ++i) fA[i] = (fg << 5) + (i << 4) + l15;
    #pragma unroll
    for (int j = 0; j < 4; ++j) dB[j] = (dg << 6) + (j << 4) + l15;

    const v2f* xb = xp + ((size_t)b << 16);

    v8f accRe[2][4], accIm[2][4];
    #pragma unroll
    for (int i = 0; i < 2; ++i)
        #pragma unroll
        for (int j = 0; j < 4; ++j) { accRe[i][j] = (v8f){}; accIm[i][j] = (v8f){}; }

    for (int kt = 0; kt < T_; kt += 4) {
        const int t2 = (kt >> 1) + hk;
        v2f ac[2], as[2], bx[4];
        #pragma unroll
        for (int i = 0; i < 2; ++i) {
            ac[i] = Wc[(t2 << 8) + fA[i]];
            as[i] = Ws[(t2 << 8) + fA[i]];
        }
        #pragma unroll
        for (int j = 0; j < 4; ++j)
            bx[j] = xb[(t2 << 8) + dB[j]];
        #pragma unroll
        for (int i = 0; i < 2; ++i)
            #pragma unroll
            for (int j = 0; j < 4; ++j) {
                accRe[i][j] = __builtin_amdgcn_wmma_f32_16x16x4_f32(
                    false, ac[i], false, bx[j], (short)0, accRe[i][j], false, false);
                accIm[i][j] = __builtin_amdgcn_wmma_f32_16x16x4_f32(
                    false, as[i], false, bx[j], (short)0, accIm[i][j], false, false);
            }
    }

    // C/D layout: VGPR v -> M = v (+8 for upper half-wave); N = lane&15.
    const size_t ob = (size_t)b << 16;
    #pragma unroll
    for (int i = 0; i < 2; ++i) {
        const int fbase = (fg << 5) + (i << 4) + (hk << 3);
        #pragma unroll
        for (int j = 0; j < 4; ++j)
            #pragma unroll
            for (int v = 0; v < 8; ++v) {
                const size_t o = ob + ((size_t)(fbase + v) << 8) + dB[j];
                Re[o] = accRe[i][j][v];
                Im[o] = -accIm[i][j][v];      // rfft imag = -sum x sin
            }
    }
}

// ---------------------------------------------------------------------------
// Stage 2: per-(b,d) top-16 over bins 1..255; write masked+scaled spectrum in
// PACKED form Pp/Qp[b][k2][d] = {v[2k2], v[2k2+1]} (B-fragment-ready), folding
// the 2/T amplitude scale and the "-Im*sin" sign.
// ---------------------------------------------------------------------------
__global__ void topk_scale(const float* __restrict__ Re, const float* __restrict__ Im,
                           float* __restrict__ Pp, float* __restrict__ Qp) {
    int tid = blockIdx.x * blockDim.x + threadIdx.x;   // 0..B*D-1
    if (tid >= B_ * D_) return;
    const int b = tid >> 8;
    const int d = tid & 255;
    const float* re = Re + ((size_t)b << 16) + d;
    const float* im = Im + ((size_t)b << 16) + d;
    float* pp = Pp + ((size_t)b << 16);
    float* qq = Qp + ((size_t)b << 16);

    float vals[K_];
    int   idxs[K_];
    for (int j = 0; j < K_; ++j) { vals[j] = -1.0f; idxs[j] = 0; }

    for (int i = 1; i < NF_; ++i) {               // bin 0 excluded (low_freq=1)
        const float r = re[i << 8];
        const float q = im[i << 8];
        const float m = r * r + q * q;
        if (m > vals[0]) {                        // keep ascending list of 16
            int p = 0;
            while (p < K_ - 1 && vals[p + 1] < m) {
                vals[p] = vals[p + 1]; idxs[p] = idxs[p + 1]; ++p;
            }
            vals[p] = m; idxs[p] = i;
        }
    }

    float sr[K_], si[K_];
    for (int j = 0; j < K_; ++j) {
        sr[j] = re[idxs[j] << 8];
        si[j] = im[idxs[j] << 8];
    }
    // zero this (b,d) column of the packed buffers, then scatter the 16 picks
    for (int k = 0; k < NF_; ++k) {
        const int a = ((k >> 1) << 9) + (d << 1) + (k & 1);
        pp[a] = 0.0f; qq[a] = 0.0f;
    }
    const float sc = 2.0f / (float)T_;
    for (int j = 0; j < K_; ++j) {
        const int i = idxs[j];
        const int a = ((i >> 1) << 9) + (d << 1) + (i & 1);
        pp[a] =  sc * sr[j];
        qq[a] = -sc * si[j];
    }
}

// ---------------------------------------------------------------------------
// Stage 3: synthesis f32 WMMA GEMM. One wave owns 32(tau) x 64(d)
// (2x4 sub-tiles). Per K=4 step: 4 A b64 + 8 B b64 loads feed 16 WMMAs.
// out[b,tau,d] = sum_f Wc[f][tau]*P + Ws[f][tau]*Q
// Grid: 32 b * 19 tgroups * 4 dgroups = 2432 waves -> 608 blocks.
// ---------------------------------------------------------------------------
__global__ __launch_bounds__(128) void synth_wmma(
    const v2f* __restrict__ Pp,     // [b][k2][d]
    const v2f* __restrict__ Qp,
    const v2f* __restrict__ Wc,     // [f2][tau]
    const v2f* __restrict__ Ws,
    float* __restrict__ out) {

    const int wave = (blockIdx.x << 2) + (threadIdx.x >> 5);   // 0..2431
    const int lane = threadIdx.x & 31;
    const int b  = wave / 76;                  // 76 = 19*4 macro-tiles/batch
    const int r  = wave - b * 76;
    const int tg = r >> 2;                     // 0..18
    const int dg = r & 3;

    const int l15 = lane & 15;
    const int hk  = lane >> 4;

    int tauA[2], dB[4];
    #pragma unroll
    for (int i = 0; i < 2; ++i) tauA[i] = (tg << 5) + (i << 4) + l15;
    #pragma unroll
    for (int j = 0; j < 4; ++j) dB[j] = (dg << 6) + (j << 4) + l15;

    const v2f* pb = Pp + ((size_t)b << 15);    // b * 128 * 256 v2f
    const v2f* qb = Qp + ((size_t)b << 15);

    v8f acc[2][4];
    #pragma unroll
    for (int i = 0; i < 2; ++i)
        #pragma unroll
        for (int j = 0; j < 4; ++j) acc[i][j] = (v8f){};

    for (int ki = 0; ki < NF_; ki += 4) {
        const int i2 = (ki >> 1) + hk;
        v2f ac[2], as[2], bp[4], bq[4];
        #pragma unroll
        for (int i = 0; i < 2; ++i) {
            ac[i] = Wc[i2 * TP_ + tauA[i]];
            as[i] = Ws[i2 * TP_ + tauA[i]];
        }
        #pragma unroll
        for (int j = 0; j < 4; ++j) {
            bp[j] = pb[(i2 << 8) + dB[j]];
            bq[j] = qb[(i2 << 8) + dB[j]];
        }
        #pragma unroll
        for (int i = 0; i < 2; ++i)
            #pragma unroll
            for (int j = 0; j < 4; ++j) {
                acc[i][j] = __builtin_amdgcn_wmma_f32_16x16x4_f32(
                    false, ac[i], false, bp[j], (short)0, acc[i][j], false, false);
                acc[i][j] = __builtin_amdgcn_wmma_f32_16x16x4_f32(
                    false, as[i], false, bq[j], (short)0, acc[i][j], false, false);
            }
    }

    const size_t ob = (size_t)b * TP_ * D_;
    #pragma unroll
    for (int i = 0; i < 2; ++i) {
        const int tbase = (tg << 5) + (i << 4) + (hk << 3);
        #pragma unroll
        for (int j = 0; j < 4; ++j)
            #pragma unroll
            for (int v = 0; v < 8; ++v)
                out[ob + (size_t)(tbase + v) * D_ + dB[j]] = acc[i][j][v];
    }
}

// ---------------------------------------------------------------------------
// Launcher. Workspace (floats):
//  WcDp[131072] WsDp[131072] WcSp[155648] WsSp[155648]
//  xp[4194304]  Re[2097152] Im[2097152] Pp[2097152] Qp[2097152]  (~53 MB)
// ---------------------------------------------------------------------------
extern "C" void kernel_launch(void* const* d_in, const int* in_sizes, int n_in,
                              void* d_out, int out_size, void* d_ws, size_t ws_size,
                              hipStream_t stream) {
    const float* x = (const float*)d_in[0];   // (32,512,256) f32
    float* out = (float*)d_out;               // (32,608,256) f32
    (void)in_sizes; (void)n_in; (void)out_size; (void)ws_size;

    float* ws = (float*)d_ws;
    v2f* WcDp = (v2f*)ws;                           // 65536 v2f
    v2f* WsDp = WcDp + 65536;
    v2f* WcSp = WsDp + 65536;                       // 77824 v2f
    v2f* WsSp = WcSp + 77824;
    v2f* xp   = WsSp + 77824;                       // 2097152 v2f
    float* Re = (float*)(xp + 2097152);             // 2097152 f each
    float* Im = Re + 2097152;
    float* Pp = Im + 2097152;
    float* Qp = Pp + 2097152;

    build_dft_tables<<<(65536 + 255) / 256, 256, 0, stream>>>(WcDp, WsDp);
    build_syn_tables<<<(77824 + 255) / 256, 256, 0, stream>>>(WcSp, WsSp);
    pack_x<<<(2097152 + 255) / 256, 256, 0, stream>>>(x, xp);

    dft_wmma<<<1024 / 4, 128, 0, stream>>>(xp, WcDp, WsDp, Re, Im);

    topk_scale<<<(B_ * D_ + 127) / 128, 128, 0, stream>>>(Re, Im, Pp, Qp);

    synth_wmma<<<2432 / 4, 128, 0, stream>>>((const v2f*)Pp, (const v2f*)Qp,
                                             WcSp, WsSp, out);
}